// Quantiser_25280177504503
// MI455X (gfx1250) — compile-verified
//
#include <hip/hip_runtime.h>
#include <hip/hip_bf16.h>

typedef __attribute__((ext_vector_type(2))) float v2f;
typedef __attribute__((ext_vector_type(8))) float v8f;

#define VQ_N     65536
#define VQ_D     64
#define VQ_V     8192
#define CHUNK    64            // codes staged to LDS per iteration
#define PAD      68            // floats per staged code row (bank-conflict-free)
#define NCHUNK   (VQ_V / CHUNK)
#define ROWS_PER_WAVE  32      // two 16-row A tiles per wave
#define ROWS_PER_BLOCK 256     // 8 waves * 32 rows
#define BUF_BYTES (CHUNK * PAD * 4)
#define CHUNK_GBYTES (CHUNK * VQ_D * 4)

// ---------------------------------------------------------------------------
// Kernel 1: e2half[v] = 0.5 * ||E_v||^2
// ---------------------------------------------------------------------------
__global__ __launch_bounds__(256) void vq_e2half_kernel(const float* __restrict__ E,
                                                        float* __restrict__ e2half) {
    int v = blockIdx.x * 256 + threadIdx.x;          // 8192 total
    const float4* row = (const float4*)(E + (size_t)v * VQ_D);
    float s = 0.f;
#pragma unroll
    for (int i = 0; i < 16; ++i) {
        float4 t = row[i];
        s += t.x * t.x + t.y * t.y + t.z * t.z + t.w * t.w;
    }
    e2half[v] = 0.5f * s;
}

// ---------------------------------------------------------------------------
// Kernel 2: per-row argmax of (x . e - ||e||^2/2) via V_WMMA_F32_16X16X4_F32.
// Each wave owns 32 rows of X (two A tiles in registers, B fragment reused for
// both -> halved LDS traffic per WMMA). E is streamed in 64-code chunks into
// double-buffered LDS with GLOBAL_LOAD_ASYNC_TO_LDS_B128 (ASYNCcnt), so the
// next chunk's DMA overlaps the current chunk's WMMA work.
// ---------------------------------------------------------------------------
__global__ __launch_bounds__(256) void vq_argmin_kernel(const float* __restrict__ x,
                                                        const float* __restrict__ E,
                                                        const float* __restrict__ e2half,
                                                        int*   __restrict__ idx_int,
                                                        float* __restrict__ idx_f) {
    __shared__ float sm[2][CHUNK * PAD];

    const int tid  = threadIdx.x;
    const int wave = tid >> 5;
    const int lane = tid & 31;
    const int l15  = lane & 15;          // N index inside tile / M row index
    const int h2   = (lane >> 4) * 2;    // K sub-offset per A/B fp32 layout
    const int rowBase = blockIdx.x * ROWS_PER_BLOCK + wave * ROWS_PER_WAVE;

    // ---- load this wave's two 16x64 A tiles into registers
    const float* xr0 = x + (size_t)(rowBase + l15) * VQ_D + h2;
    const float* xr1 = x + (size_t)(rowBase + 16 + l15) * VQ_D + h2;
    v2f a0[16], a1[16];
#pragma unroll
    for (int t = 0; t < 16; ++t) { a0[t] = *(const v2f*)(xr0 + 4 * t);
                                   a1[t] = *(const v2f*)(xr1 + 4 * t); }

    float best[16];
    unsigned int bidx[16];
#pragma unroll
    for (int r = 0; r < 16; ++r) { best[r] = -3.4e38f; bidx[r] = 0u; }

    // ---- async staging setup: each thread copies 64 contiguous bytes
    const int code = tid >> 2;           // 0..63
    const int part = tid & 3;            // 0..3
    const char* gsrc = (const char*)(E + (size_t)code * VQ_D + part * 16);
    const unsigned ldsoff = (unsigned)(size_t)(const void*)&sm[0][0]
                          + (unsigned)(code * PAD + part * 16) * 4u;

#define STAGE(bufsel, gptr)                                                     \
    do {                                                                        \
        unsigned _la = ldsoff + (unsigned)(bufsel) * BUF_BYTES;                 \
        asm volatile(                                                           \
            "global_load_async_to_lds_b128 %0, %1, off\n\t"                     \
            "global_load_async_to_lds_b128 %0, %1, off offset:16\n\t"           \
            "global_load_async_to_lds_b128 %0, %1, off offset:32\n\t"           \
            "global_load_async_to_lds_b128 %0, %1, off offset:48"               \
            :: "v"(_la), "v"(gptr) : "memory");                                 \
    } while (0)

    // prologue: stage chunk 0
    STAGE(0, gsrc);
    asm volatile("s_wait_asynccnt 0x0" ::: "memory");
    __syncthreads();

    for (int c = 0; c < NCHUNK; ++c) {
        const int buf = c & 1;
        if (c + 1 < NCHUNK)
            STAGE(buf ^ 1, gsrc + (size_t)(c + 1) * CHUNK_GBYTES);

        // prefetch the 4 bias values for this chunk (complete under WMMA work)
        float e2v[4];
#pragma unroll
        for (int tile = 0; tile < 4; ++tile)
            e2v[tile] = e2half[c * CHUNK + tile * 16 + l15];

#pragma unroll
        for (int tile = 0; tile < 4; ++tile) {
            const float* bbase = &sm[buf][(tile * 16 + l15) * PAD + h2];
            v8f acc0a = {}, acc0b = {};   // row tile 0, even/odd k
            v8f acc1a = {}, acc1b = {};   // row tile 1, even/odd k
#pragma unroll
            for (int t = 0; t < 16; t += 2) {
                v2f b0 = *(const v2f*)(bbase + 4 * t);
                acc0a = __builtin_amdgcn_wmma_f32_16x16x4_f32(
                    false, a0[t],     false, b0, (short)0, acc0a, false, false);
                acc1a = __builtin_amdgcn_wmma_f32_16x16x4_f32(
                    false, a1[t],     false, b0, (short)0, acc1a, false, false);
                v2f b1 = *(const v2f*)(bbase + 4 * t + 4);
                acc0b = __builtin_amdgcn_wmma_f32_16x16x4_f32(
                    false, a0[t + 1], false, b1, (short)0, acc0b, false, false);
                acc1b = __builtin_amdgcn_wmma_f32_16x16x4_f32(
                    false, a1[t + 1], false, b1, (short)0, acc1b, false, false);
            }
            v8f s0 = acc0a + acc0b;
            v8f s1 = acc1a + acc1b;

            const unsigned int codeid = (unsigned int)(c * CHUNK + tile * 16 + l15);
            const float e2 = e2v[tile];
#pragma unroll
            for (int r = 0; r < 8; ++r) {
                float sc0 = s0[r] - e2;
                if (sc0 > best[r]) { best[r] = sc0; bidx[r] = codeid; }
                float sc1 = s1[r] - e2;
                if (sc1 > best[8 + r]) { best[8 + r] = sc1; bidx[8 + r] = codeid; }
            }
        }

        asm volatile("s_wait_asynccnt 0x0" ::: "memory");
        __syncthreads();
    }
#undef STAGE

    // ---- reduce across the 16 lanes holding each row (halves stay separate)
#pragma unroll
    for (int rt = 0; rt < 2; ++rt) {
#pragma unroll
        for (int r = 0; r < 8; ++r) {
            float bs = best[rt * 8 + r];
            unsigned int bi = bidx[rt * 8 + r];
#pragma unroll
            for (int m = 1; m < 16; m <<= 1) {
                float        os = __shfl_xor(bs, m, 32);
                unsigned int oi = (unsigned int)__shfl_xor((int)bi, m, 32);
                if (os > bs || (os == bs && oi < bi)) { bs = os; bi = oi; }
            }
            if (l15 == 0) {
                int row = rowBase + rt * 16 + r + (lane >> 4) * 8;  // lanes 0 / 16
                idx_int[row] = (int)bi;
                idx_f[row]   = (float)bi;
            }
        }
    }
}

// ---------------------------------------------------------------------------
// Kernel 3: quantised = E[idx]; loss += 1.25 * mean((q - x)^2)   (float4 wide)
// ---------------------------------------------------------------------------
__global__ __launch_bounds__(256) void vq_gather_loss_kernel(const float* __restrict__ x,
                                                             const float* __restrict__ E,
                                                             const int* __restrict__ idx_int,
                                                             float* __restrict__ q_out,
                                                             float* __restrict__ loss) {
    const float SCALE = 1.25f / ((float)VQ_N * (float)VQ_D);
    int gid = blockIdx.x * 256 + threadIdx.x;   // N*16 total
    int row = gid >> 4;
    int d4  = (gid & 15) * 4;
    float4 q  = *(const float4*)(E + (size_t)idx_int[row] * VQ_D + d4);
    float4 xv = *(const float4*)(x + (size_t)row * VQ_D + d4);
    *(float4*)(q_out + (size_t)row * VQ_D + d4) = q;
    float dx = q.x - xv.x, dy = q.y - xv.y, dz = q.z - xv.z, dw = q.w - xv.w;
    float p = dx * dx + dy * dy + dz * dz + dw * dw;

#pragma unroll
    for (int m = 16; m > 0; m >>= 1) p += __shfl_xor(p, m, 32);

    __shared__ float ws[8];
    if ((threadIdx.x & 31) == 0) ws[threadIdx.x >> 5] = p;
    __syncthreads();
    if (threadIdx.x < 8) {
        float v = ws[threadIdx.x];
#pragma unroll
        for (int m = 4; m > 0; m >>= 1) v += __shfl_xor(v, m, 32);
        if (threadIdx.x == 0) atomicAdd(loss, v * SCALE);
    }
}

// ---------------------------------------------------------------------------
extern "C" void kernel_launch(void* const* d_in, const int* in_sizes, int n_in,
                              void* d_out, int out_size, void* d_ws, size_t ws_size,
                              hipStream_t stream) {
    const float* x = (const float*)d_in[0];          // [N, 64]
    const float* E = (const float*)d_in[1];          // [V, 64]

    float* q_out  = (float*)d_out;                   // N*64 floats
    float* loss   = q_out + (size_t)VQ_N * VQ_D;     // 1 float
    float* idx_f  = loss + 1;                        // N floats

    float* e2half  = (float*)d_ws;                               // V floats
    int*   idx_int = (int*)((char*)d_ws + VQ_V * sizeof(float)); // N ints

    vq_e2half_kernel<<<VQ_V / 256, 256, 0, stream>>>(E, e2half);
    vq_argmin_kernel<<<VQ_N / ROWS_PER_BLOCK, 256, 0, stream>>>(x, E, e2half,
                                                                idx_int, idx_f);
    hipMemsetAsync(loss, 0, sizeof(float), stream);
    vq_gather_loss_kernel<<<(VQ_N * VQ_D / 4) / 256, 256, 0, stream>>>(x, E, idx_int,
                                                                      q_out, loss);
}